// ConjugateGradientModel_3083786518605
// MI455X (gfx1250) — compile-verified
//
#include <hip/hip_runtime.h>

typedef float v2f __attribute__((ext_vector_type(2)));
typedef float v8f __attribute__((ext_vector_type(8)));

#define BATCH 4
#define H 256
#define W 256
#define Z 64
#define NB (H * W * Z)               // 4,194,304 per batch
#define NTOT (BATCH * NB)            // 16,777,216
#define EPS 1e-6f
#define ITERS 8
#define ROWSTRIDE 68                 // 64 + 4 pad floats; 272B = 17*16B (keeps float4 align, kills bank conflicts)

// ---------------------------------------------------------------------------
// Stencil kernel: one wave handles a tile of 16 w-positions x 64 z at fixed (b,h).
// Center tile staged into LDS via GLOBAL_LOAD_ASYNC_TO_LDS_B128 (ASYNCcnt path,
// no VGPR round-trip). Z-coupling (6*x - x[z-1] - x[z+1]) via
// V_WMMA_F32_16X16X4_F32 against a constant 16x16 tridiagonal B; H/W neighbors
// + chunk-edge z terms go into the accumulator C.
// INIT:  r = b - stencil(x); p = r; x_out = x; partial += r*r
// STEP:  w = stencil(p);              partial += p*w
// ---------------------------------------------------------------------------
template <bool INIT>
__global__ __launch_bounds__(512) void stencil_kernel(
    const float* __restrict__ pin,   // x (INIT) or p (STEP)
    const float* __restrict__ bin,   // b (INIT only)
    float* __restrict__ wout,        // w (STEP only)
    float* __restrict__ rout,        // r (INIT only)
    float* __restrict__ pout,        // p (INIT only)
    float* __restrict__ xout,        // x working copy = d_out+1 (INIT only)
    float* __restrict__ partial)     // per-block partial dot, [B*H]
{
  __shared__ alignas(16) float lds[16 * 16 * ROWSTRIDE];
  __shared__ float red[16];

  const int tid  = threadIdx.x;
  const int wv   = tid >> 5;
  const int lane = tid & 31;
  const int h    = blockIdx.x;
  const int b    = blockIdx.y;
  const int w0   = wv * 16;

  float* tile = lds + wv * (16 * ROWSTRIDE);
  const size_t rowbase = (((size_t)b * H + h) * W + w0) * Z;

  // Async (direct-to-LDS) coalesced fill of the 16x64 center tile.
#pragma unroll
  for (int i = 0; i < 8; ++i) {
    const int idx = lane + 32 * i;       // 0..255 float4 slots
    const int m = idx >> 4;              // row 0..15
    const int q = idx & 15;              // z/4
    const float* gsrc = pin + rowbase + (size_t)m * Z + 4 * q;
    const unsigned ldsOff = (unsigned)(size_t)(tile + m * ROWSTRIDE + 4 * q);
    asm volatile("global_load_async_to_lds_b128 %0, %1, off"
                 :: "v"(ldsOff), "v"(gsrc)
                 : "memory");
  }
  asm volatile("s_wait_asynccnt 0x0" ::: "memory");
  __syncthreads();

  const int hi = lane >> 4;   // lane-half
  const int n  = lane & 15;   // column index (z within chunk) for B/C/D layouts

  // Constant B fragments: T[i][n] = 6 (i==n), -1 (|i-n|==1), else 0.
  // B layout (4x16 f32): lane holds rows k_local = 2*hi + j, col = lane&15.
  v2f bfrag[4];
#pragma unroll
  for (int ks = 0; ks < 4; ++ks) {
#pragma unroll
    for (int j = 0; j < 2; ++j) {
      const int i = 4 * ks + 2 * hi + j;
      const int d = i - n;
      bfrag[ks][j] = (d == 0) ? 6.0f : ((d == 1 || d == -1) ? -1.0f : 0.0f);
    }
  }

  float part = 0.0f;

  for (int c = 0; c < 4; ++c) {
    const int zc = c * 16;

    // C init: -(north + south + west + east + z chunk-edge halo)
    v8f acc;
#pragma unroll
    for (int v = 0; v < 8; ++v) {
      const int m = v + 8 * hi;
      const int z = zc + n;
      const int wq = w0 + m;
      const size_t cidx = rowbase + (size_t)m * Z + z;
      float s = 0.0f;
      if (h > 0)      s += pin[cidx - (size_t)(W * Z)];
      if (h < H - 1)  s += pin[cidx + (size_t)(W * Z)];
      if (wq > 0)     s += pin[cidx - Z];
      if (wq < W - 1) s += pin[cidx + Z];
      if (n == 0 && c > 0)  s += tile[m * ROWSTRIDE + zc - 1];
      if (n == 15 && c < 3) s += tile[m * ROWSTRIDE + zc + 16];
      acc[v] = -s;
    }

    // A layout (16x4 f32): lane holds A[lane&15][2*hi + j].
#pragma unroll
    for (int ks = 0; ks < 4; ++ks) {
      const int zA = zc + 4 * ks + 2 * hi;
      const float* ap = tile + (lane & 15) * ROWSTRIDE + zA;
      v2f a;
      a[0] = ap[0];
      a[1] = ap[1];
      acc = __builtin_amdgcn_wmma_f32_16x16x4_f32(false, a, false, bfrag[ks],
                                                  (short)0, acc, false, false);
    }

    // Emit D and accumulate the fused dot product.
#pragma unroll
    for (int v = 0; v < 8; ++v) {
      const int m = v + 8 * hi;
      const int z = zc + n;
      const size_t cidx = rowbase + (size_t)m * Z + z;
      const float cen = tile[m * ROWSTRIDE + z];
      if (INIT) {
        const float rv = bin[cidx] - acc[v];
        rout[cidx] = rv;
        pout[cidx] = rv;
        xout[cidx] = cen;
        part += rv * rv;
      } else {
        wout[cidx] = acc[v];
        part += acc[v] * cen;
      }
    }
  }

  // Deterministic wave + block reduction.
#pragma unroll
  for (int off = 16; off > 0; off >>= 1) part += __shfl_down(part, off, 32);
  if (lane == 0) red[wv] = part;
  __syncthreads();
  if (tid == 0) {
    float s = 0.0f;
#pragma unroll
    for (int i = 0; i < 16; ++i) s += red[i];
    partial[b * H + h] = s;
  }
}

// rho_b = sum(partial[b*256 .. b*256+255])  (initial rho)
__global__ __launch_bounds__(256) void reduce_rho_init_kernel(
    const float* __restrict__ part, float* __restrict__ scal)
{
  __shared__ float sm[256];
  const int t = threadIdx.x;
  for (int b = 0; b < BATCH; ++b) {
    sm[t] = part[b * 256 + t];
    __syncthreads();
    for (int s = 128; s > 0; s >>= 1) {
      if (t < s) sm[t] += sm[t + s];
      __syncthreads();
    }
    if (t == 0) scal[b] = sm[0];
    __syncthreads();
  }
}

// alpha_b = rho_b / (eta_b + eps)
__global__ __launch_bounds__(256) void reduce_eta_kernel(
    const float* __restrict__ part, float* __restrict__ scal)
{
  __shared__ float sm[256];
  const int t = threadIdx.x;
  for (int b = 0; b < BATCH; ++b) {
    sm[t] = part[b * 256 + t];
    __syncthreads();
    for (int s = 128; s > 0; s >>= 1) {
      if (t < s) sm[t] += sm[t + s];
      __syncthreads();
    }
    if (t == 0) scal[4 + b] = scal[b] / (sm[0] + EPS);
    __syncthreads();
  }
}

// rho_next; beta_b = rho_next/(rho_b+eps); rho_b = rho_next
__global__ __launch_bounds__(256) void reduce_rho_kernel(
    const float* __restrict__ part, float* __restrict__ scal)
{
  __shared__ float sm[256];
  const int t = threadIdx.x;
  for (int b = 0; b < BATCH; ++b) {
    const float* pb = part + b * 1024;
    sm[t] = pb[t] + pb[t + 256] + pb[t + 512] + pb[t + 768];
    __syncthreads();
    for (int s = 128; s > 0; s >>= 1) {
      if (t < s) sm[t] += sm[t + s];
      __syncthreads();
    }
    if (t == 0) {
      const float rn = sm[0];
      scal[8 + b] = rn / (scal[b] + EPS);
      scal[b] = rn;
    }
    __syncthreads();
  }
}

// x += alpha*p; r -= alpha*w; partial += r_new*r_new   (x = d_out+1, scalar access)
__global__ __launch_bounds__(256) void update_kernel(
    float* __restrict__ x, float* __restrict__ r,
    const float* __restrict__ p, const float* __restrict__ w,
    const float* __restrict__ scal, float* __restrict__ partial)
{
  const int b = blockIdx.y;
  const float alpha = scal[4 + b];
  const size_t base = (size_t)b * NB + (size_t)blockIdx.x * 4096;
  float part = 0.0f;
#pragma unroll
  for (int q = 0; q < 4; ++q) {
    const size_t i = base + ((size_t)q * 256 + threadIdx.x) * 4;
    const float4 pv = *reinterpret_cast<const float4*>(p + i);
    const float4 wv4 = *reinterpret_cast<const float4*>(w + i);
    float4 rv = *reinterpret_cast<float4*>(r + i);
    x[i + 0] += alpha * pv.x;
    x[i + 1] += alpha * pv.y;
    x[i + 2] += alpha * pv.z;
    x[i + 3] += alpha * pv.w;
    rv.x -= alpha * wv4.x;
    rv.y -= alpha * wv4.y;
    rv.z -= alpha * wv4.z;
    rv.w -= alpha * wv4.w;
    *reinterpret_cast<float4*>(r + i) = rv;
    part += rv.x * rv.x + rv.y * rv.y + rv.z * rv.z + rv.w * rv.w;
  }
#pragma unroll
  for (int off = 16; off > 0; off >>= 1) part += __shfl_down(part, off, 32);
  __shared__ float red[8];
  const int wv = threadIdx.x >> 5;
  if ((threadIdx.x & 31) == 0) red[wv] = part;
  __syncthreads();
  if (threadIdx.x == 0) {
    float s = 0.0f;
#pragma unroll
    for (int i = 0; i < 8; ++i) s += red[i];
    partial[b * 1024 + blockIdx.x] = s;
  }
}

// p = r + beta*p
__global__ __launch_bounds__(256) void pupdate_kernel(
    float* __restrict__ p, const float* __restrict__ r, const float* __restrict__ scal)
{
  const int b = blockIdx.y;
  const float beta = scal[8 + b];
  const size_t base = (size_t)b * NB + (size_t)blockIdx.x * 4096;
#pragma unroll
  for (int q = 0; q < 4; ++q) {
    const size_t i = base + ((size_t)q * 256 + threadIdx.x) * 4;
    const float4 rv = *reinterpret_cast<const float4*>(r + i);
    float4 pv = *reinterpret_cast<const float4*>(p + i);
    pv.x = rv.x + beta * pv.x;
    pv.y = rv.y + beta * pv.y;
    pv.z = rv.z + beta * pv.z;
    pv.w = rv.w + beta * pv.w;
    *reinterpret_cast<float4*>(p + i) = pv;
  }
}

// loss/max-error partials over all N
__global__ __launch_bounds__(256) void final_kernel(
    const float* __restrict__ x, const float* __restrict__ ref,
    float* __restrict__ lossPart, float* __restrict__ maxPart)
{
  const size_t base = (size_t)blockIdx.x * 4096;
  float s = 0.0f, mx = 0.0f;
#pragma unroll
  for (int q = 0; q < 16; ++q) {
    const size_t i = base + (size_t)q * 256 + threadIdx.x;
    const float d = x[i] - ref[i];
    s += d * d;
    mx = fmaxf(mx, fabsf(d));
  }
#pragma unroll
  for (int off = 16; off > 0; off >>= 1) {
    s += __shfl_down(s, off, 32);
    mx = fmaxf(mx, __shfl_down(mx, off, 32));
  }
  __shared__ float reds[8], redm[8];
  const int wv = threadIdx.x >> 5;
  if ((threadIdx.x & 31) == 0) { reds[wv] = s; redm[wv] = mx; }
  __syncthreads();
  if (threadIdx.x == 0) {
    float ss = 0.0f, mm = 0.0f;
#pragma unroll
    for (int i = 0; i < 8; ++i) { ss += reds[i]; mm = fmaxf(mm, redm[i]); }
    lossPart[blockIdx.x] = ss;
    maxPart[blockIdx.x] = mm;
  }
}

__global__ __launch_bounds__(256) void final2_kernel(
    const float* __restrict__ lossPart, const float* __restrict__ maxPart,
    const float* __restrict__ scal, float* __restrict__ out)
{
  __shared__ float sm[256], mm[256];
  const int t = threadIdx.x;
  float s = 0.0f, m = 0.0f;
#pragma unroll
  for (int k = 0; k < 16; ++k) {
    s += lossPart[k * 256 + t];
    m = fmaxf(m, maxPart[k * 256 + t]);
  }
  sm[t] = s; mm[t] = m;
  __syncthreads();
  for (int st = 128; st > 0; st >>= 1) {
    if (t < st) { sm[t] += sm[t + st]; mm[t] = fmaxf(mm[t], mm[t + st]); }
    __syncthreads();
  }
  if (t == 0) {
    out[0] = sm[0] / (float)NTOT;
    out[1 + NTOT] = mm[0];
    out[2 + NTOT] = 0.25f * (scal[0] + scal[1] + scal[2] + scal[3]);
  }
}

extern "C" void kernel_launch(void* const* d_in, const int* in_sizes, int n_in,
                              void* d_out, int out_size, void* d_ws, size_t ws_size,
                              hipStream_t stream) {
  (void)in_sizes; (void)n_in; (void)out_size; (void)ws_size;
  const float* x_in = (const float*)d_in[0];
  const float* b_in = (const float*)d_in[1];
  const float* ref  = (const float*)d_in[2];

  float* out = (float*)d_out;
  float* xp  = out + 1;                 // working x lives in the output buffer

  float* ws = (float*)d_ws;
  const size_t Nf = (size_t)NTOT;
  float* r    = ws;                     // N
  float* p    = ws + Nf;                // N
  float* w    = ws + 2 * Nf;            // N
  float* scal = ws + 3 * Nf;            // [0..3]=rho [4..7]=alpha [8..11]=beta
  float* P1   = scal + 16;              // 1024 stencil partials
  float* P2   = P1 + 1024;              // 4096 update partials
  float* P3   = P2 + 4096;              // 4096 loss partials
  float* P4   = P3 + 4096;              // 4096 max partials

  const dim3 gs(H, BATCH);              // stencil grid: (h, b), 16 waves/block
  const dim3 gu(1024, BATCH);           // elementwise grid per batch

  // r = b - A x ; p = r ; x working copy ; rho partials
  stencil_kernel<true><<<gs, 512, 0, stream>>>(x_in, b_in, nullptr, r, p, xp, P1);
  reduce_rho_init_kernel<<<1, 256, 0, stream>>>(P1, scal);

  for (int it = 0; it < ITERS; ++it) {
    stencil_kernel<false><<<gs, 512, 0, stream>>>(p, nullptr, w, nullptr, nullptr, nullptr, P1);
    reduce_eta_kernel<<<1, 256, 0, stream>>>(P1, scal);
    update_kernel<<<gu, 256, 0, stream>>>(xp, r, p, w, scal, P2);
    reduce_rho_kernel<<<1, 256, 0, stream>>>(P2, scal);
    pupdate_kernel<<<gu, 256, 0, stream>>>(p, r, scal);
  }

  final_kernel<<<4096, 256, 0, stream>>>(xp, ref, P3, P4);
  final2_kernel<<<1, 256, 0, stream>>>(P3, P4, scal, out);
}